// RandPool_36739150250678
// MI455X (gfx1250) — compile-verified
//
#include <hip/hip_runtime.h>

typedef __attribute__((ext_vector_type(2))) float v2f;
typedef __attribute__((ext_vector_type(8))) float v8f;

#define NB     16
#define NPTS   4096
#define NNODE  1024
#define NFEAT  64
#define KNN    16
#define WPB    8          // waves per block (256 threads)
#define PC     (NB*3*NNODE)          // 49152 floats per pool_coords copy
#define PFBASE (2*PC)                // offset of pool_feats in d_out

// ---------------------------------------------------------------------------
// Isolated WMMA tile: D = A x B (+0), scatter lane's 8 result floats to LDS.
// __noinline__ so ALL vector-tuple live ranges (v2f operands, v8f result)
// exist only inside this tiny single-block function. (This helper compiled
// cleanly in previous rounds; the ICE was in the caller's scalar code.)
// ---------------------------------------------------------------------------
__device__ __noinline__ void wmma_dist_tile(float a0, float a1,
                                            float b0, float b1,
                                            float* wp) {
    v2f a; a.x = a0; a.y = a1;
    v2f b; b.x = b0; b.y = b1;
    v8f c = {};
    v8f d = __builtin_amdgcn_wmma_f32_16x16x4_f32(
        /*neg_a=*/false, a, /*neg_b=*/false, b,
        /*c_mod=*/(short)0, c, /*reuse_a=*/false, /*reuse_b=*/false);
#pragma unroll
    for (int r = 0; r < 8; ++r) wp[r] = d[r];
}

// ---------------------------------------------------------------------------
// Trivial copies: pool_coords (twice) and pool_feats channels 0..63.
// ---------------------------------------------------------------------------
__global__ __launch_bounds__(256)
void randpool_copy_kernel(const float* __restrict__ coords,
                          const float* __restrict__ feats,
                          float* __restrict__ out) {
    int i = blockIdx.x * blockDim.x + threadIdx.x;
    const int PFH = NB * NFEAT * NNODE;             // lower-half pool_feats
    if (i < 2 * PC) {
        int j = (i < PC) ? i : (i - PC);
        int b = j / (3 * NNODE);
        int r = j % (3 * NNODE);
        int c = r / NNODE;
        int m = r % NNODE;
        out[i] = coords[(b * 3 + c) * NPTS + m];
    } else {
        int j = i - 2 * PC;
        if (j < PFH) {
            int b = j / (NFEAT * NNODE);
            int r = j % (NFEAT * NNODE);
            int c = r / NNODE;
            int m = r % NNODE;
            out[PFBASE + (b * 128 + c) * NNODE + m] =
                feats[(b * NFEAT + c) * NPTS + m];
        }
    }
}

// ---------------------------------------------------------------------------
// kNN (top-16 by squared L2) + feature max-pool.
// One wave handles 16 nodes. Distance tiles via V_WMMA_F32_16X16X4_F32 with
// A[m,:] = (-2qx,-2qy,-2qz, 1), B[:,n] = (px,py,pz,|p|^2)  =>
// D[m,n] = |p|^2 - 2 q.p  (rank-equivalent to d^2 per node row).
// Top-16 kept as a SORTED ascending list in registers; insertion is a
// branchless bubble pass (cndmask chains only — avoids the EXEC-chain
// pattern that ICE'd the Register Coalescer).
// ---------------------------------------------------------------------------
__global__ __launch_bounds__(256)
void randpool_knn_kernel(const float* __restrict__ coords,
                         const float* __restrict__ feats,
                         float* __restrict__ out) {
    __shared__ float s_tile [WPB * 256];   // 16x16 distance tile per wave
    __shared__ float s_dumpd[WPB * 256];   // partner-lane top-16 dists
    __shared__ int   s_dumpi[WPB * 256];   // partner-lane top-16 idxs
    __shared__ int   s_fidx [WPB * 256];   // final 16 idx per node

    const int b     = blockIdx.y;
    const int wave  = threadIdx.x >> 5;
    const int lid   = threadIdx.x & 31;
    const int half  = lid >> 4;            // 0: K=0,1 lanes ; 1: K=2,3 lanes
    const int l     = lid & 15;
    const int mbase = blockIdx.x * (WPB * 16) + wave * 16;

    float* tile  = s_tile  + wave * 256;
    float* dumpd = s_dumpd + wave * 256;
    int*   dumpi = s_dumpi + wave * 256;
    int*   fidx  = s_fidx  + wave * 256;

    const float* cb = coords + b * 3 * NPTS;

    // --- A matrix operands (node tile), loaded once; branchless lane split ---
    int m = mbase + l;
    float qx = cb[m], qy = cb[NPTS + m], qz = cb[2 * NPTS + m];
    float a0 = half ? (-2.0f * qz) : (-2.0f * qx);
    float a1 = half ? 1.0f         : (-2.0f * qy);

    // --- register-resident sorted top-16 (ascending; [15] = current max) ---
    float kd[KNN];
    int   ki[KNN];
#pragma unroll
    for (int i = 0; i < KNN; ++i) { kd[i] = 3.0e38f; ki[i] = 0; }

    const int myNode = l;                       // node this lane selects for
    const int hiBase = (myNode >= 8) ? 16 : 0;  // LDS tile addressing
    const int rowOff = (myNode & 7);
    const int c0     = half * 8;                // lane pair splits 16 columns

    float* wp = tile + (half * 16 + l) * 8;     // this lane's store slot

    for (int t = 0; t < 256; ++t) {
        // --- B matrix operands (point tile), branchless lane split ---
        int n = t * 16 + l;
        float px = cb[n], py = cb[NPTS + n], pz = cb[2 * NPTS + n];
        float pn2 = fmaf(px, px, fmaf(py, py, pz * pz));
        float b0 = half ? pz  : px;
        float b1 = half ? pn2 : py;

        // D tile: value for (node = half*8+r, col = l) lands at
        // tile[(half*16 + l)*8 + r]  (contiguous 8 floats per lane)
        wmma_dist_tile(a0, a1, b0, b1, wp);
        __syncthreads();

        // gather 8 candidate columns for my node, branchless sorted insert
#pragma unroll
        for (int j = 0; j < 8; ++j) {
            int col   = c0 + j;
            float dv  = tile[(hiBase + col) * 8 + rowOff];
            int   idx = t * 16 + col;
            if (dv < kd[KNN - 1]) {
                float cd = dv; int ci = idx;        // carry bubbles upward
#pragma unroll
                for (int i = 0; i < KNN; ++i) {
                    bool  sw = kd[i] > cd;
                    float od = kd[i]; int oi = ki[i];
                    kd[i] = sw ? cd : od;
                    ki[i] = sw ? ci : oi;
                    cd    = sw ? od : cd;
                    ci    = sw ? oi : ci;
                }
            }
        }
        __syncthreads();
    }

    // --- merge lane pair (l) and (l+16) -> node l's final top-16 ---
    if (half == 1) {
#pragma unroll
        for (int i = 0; i < KNN; ++i) {
            dumpd[l * KNN + i] = kd[i];
            dumpi[l * KNN + i] = ki[i];
        }
    }
    __syncthreads();
    if (half == 0) {
#pragma unroll
        for (int i = 0; i < KNN; ++i) {
            float dv  = dumpd[l * KNN + i];
            int   idx = dumpi[l * KNN + i];
            if (dv < kd[KNN - 1]) {
                float cd = dv; int ci = idx;
#pragma unroll
                for (int k = 0; k < KNN; ++k) {
                    bool  sw = kd[k] > cd;
                    float od = kd[k]; int oi = ki[k];
                    kd[k] = sw ? cd : od;
                    ki[k] = sw ? ci : oi;
                    cd    = sw ? od : cd;
                    ci    = sw ? oi : ci;
                }
            }
        }
#pragma unroll
        for (int i = 0; i < KNN; ++i) fidx[l * KNN + i] = ki[i];
    }
    __syncthreads();

    // --- feature gather-max: 2 channels per lane x 16 nodes ---
    const float* fb    = feats + b * NFEAT * NPTS;
    float*       outpf = out + PFBASE + b * 128 * NNODE;
#pragma unroll
    for (int c2 = 0; c2 < 2; ++c2) {
        int c = lid * 2 + c2;                     // 0..63
        const float* fc = fb + c * NPTS;
        for (int mm = 0; mm < 16; ++mm) {
            float mx = -3.0e38f;
#pragma unroll
            for (int j = 0; j < KNN; ++j)
                mx = fmaxf(mx, fc[fidx[mm * KNN + j]]);
            outpf[(NFEAT + c) * NNODE + (mbase + mm)] = mx;
        }
    }
}

// ---------------------------------------------------------------------------
extern "C" void kernel_launch(void* const* d_in, const int* in_sizes, int n_in,
                              void* d_out, int out_size, void* d_ws, size_t ws_size,
                              hipStream_t stream) {
    (void)in_sizes; (void)n_in; (void)out_size; (void)d_ws; (void)ws_size;
    const float* coords = (const float*)d_in[0];
    const float* feats  = (const float*)d_in[1];
    float*       out    = (float*)d_out;

    // copies: 2*49152 + 16*64*1024 = 1,146,880 elements
    int copyN = 2 * PC + NB * NFEAT * NNODE;
    randpool_copy_kernel<<<(copyN + 255) / 256, 256, 0, stream>>>(coords, feats, out);

    // kNN + aggregate: 8 node-groups x 16 batches, 256 threads (8 waves)
    randpool_knn_kernel<<<dim3(NNODE / (WPB * 16), NB), 256, 0, stream>>>(coords, feats, out);
}